// MedianFilter1D_82566451298921
// MI455X (gfx1250) — compile-verified
//
#include <hip/hip_runtime.h>

// Median-of-9 1D filter, replicate padding. B=8, C=32, L=131072, fp32.
// Memory-bound: ~268 MB traffic -> ~11.5us floor at 23.3 TB/s.
// Strategy: per-wave private LDS staging via gfx1250 async global->LDS loads
// (no barriers needed: each wave32 stages and consumes only its own region),
// b128-vectorized global loads/stores, 8 outputs per lane, Paeth 19-CE
// median-of-9 sorting network in registers (exact selection, matches
// jnp.sort(win,axis=0)[4] bitwise).

#define HALO   4                     // (9-1)/2
#define LANES  32                    // wave32
#define OPL    8                     // outputs per lane
#define OPW    (LANES * OPL)         // 256 outputs per wave
#define NWAVES 8
#define OPB    (OPW * NWAVES)        // 2048 outputs per block
#define REGION (OPW + 2 * HALO)      // 264 staged floats per wave

#define __AS1 __attribute__((address_space(1)))
#define __AS3 __attribute__((address_space(3)))

typedef int v4i __attribute__((vector_size(4 * sizeof(int))));

#if defined(__AMDGCN__) && \
    __has_builtin(__builtin_amdgcn_global_load_async_to_lds_b128) && \
    __has_builtin(__builtin_amdgcn_global_load_async_to_lds_b32)
#define USE_ASYNC_LDS 1
#else
#define USE_ASYNC_LDS 0
#endif

__device__ __forceinline__ void ce(float& a, float& b) {
    float t = fminf(a, b);
    b = fmaxf(a, b);
    a = t;
}

// Paeth's 19-compare-exchange median-of-9 network; returns exact middle.
__device__ __forceinline__ float med9(float p0, float p1, float p2,
                                      float p3, float p4, float p5,
                                      float p6, float p7, float p8) {
    ce(p1, p2); ce(p4, p5); ce(p7, p8);
    ce(p0, p1); ce(p3, p4); ce(p6, p7);
    ce(p1, p2); ce(p4, p5); ce(p7, p8);
    ce(p0, p3); ce(p5, p8); ce(p4, p7);
    ce(p3, p6); ce(p1, p4); ce(p2, p5);
    ce(p4, p7); ce(p4, p2); ce(p6, p4);
    ce(p4, p2);
    return p4;
}

__global__ __launch_bounds__(LANES * NWAVES)
void MedianFilter1D_kernel(const float* __restrict__ x,
                           float* __restrict__ y,
                           int L) {
    const int lane = threadIdx.x & (LANES - 1);
    const int wave = threadIdx.x >> 5;
    const long long rowOff = (long long)blockIdx.y * (long long)L;
    const int s = blockIdx.x * OPB + wave * OPW;   // first output of this wave
    const float* gsrc = x + rowOff;

    __shared__ float smem[NWAVES][REGION];
    float* sreg = &smem[wave][0];

    const bool interior = (s >= HALO) && (s + OPW + HALO <= L);

#if USE_ASYNC_LDS
    if (interior) {
        const float* g0 = gsrc + (s - HALO);       // 16B aligned: s%256==0
        __builtin_amdgcn_global_load_async_to_lds_b128(
            (__AS1 v4i*)(g0 + lane * 4),
            (__AS3 v4i*)(sreg + lane * 4), 0, 0);
        __builtin_amdgcn_global_load_async_to_lds_b128(
            (__AS1 v4i*)(g0 + 128 + lane * 4),
            (__AS3 v4i*)(sreg + 128 + lane * 4), 0, 0);
        if (lane < REGION - 256) {                 // tail: 8 floats
            __builtin_amdgcn_global_load_async_to_lds_b32(
                (__AS1 int*)(g0 + 256 + lane),
                (__AS3 int*)(sreg + 256 + lane), 0, 0);
        }
    } else {
        // Row edge: per-lane clamped indices implement replicate padding.
#pragma unroll
        for (int r = 0; r < (REGION + LANES - 1) / LANES; ++r) {
            int slot = r * LANES + lane;
            if (slot < REGION) {
                int gi = s - HALO + slot;
                gi = gi < 0 ? 0 : (gi > L - 1 ? L - 1 : gi);
                __builtin_amdgcn_global_load_async_to_lds_b32(
                    (__AS1 int*)(gsrc + gi),
                    (__AS3 int*)(sreg + slot), 0, 0);
            }
        }
    }
#if __has_builtin(__builtin_amdgcn_s_wait_asynccnt)
    __builtin_amdgcn_s_wait_asynccnt(0);
#else
    asm volatile("s_wait_asynccnt 0" ::: "memory");
#endif
#else
    // Fallback: plain global load -> LDS store (same per-wave region,
    // still no barrier needed; compiler inserts the load/ds waits).
    (void)interior;
#pragma unroll
    for (int r = 0; r < (REGION + LANES - 1) / LANES; ++r) {
        int slot = r * LANES + lane;
        if (slot < REGION) {
            int gi = s - HALO + slot;
            gi = gi < 0 ? 0 : (gi > L - 1 ? L - 1 : gi);
            sreg[slot] = gsrc[gi];
        }
    }
#endif

    // 16 window floats for 8 overlapping outputs: slots [lane*8, lane*8+16).
    const float4* sv = (const float4*)(sreg + lane * OPL);  // 16B aligned
    float4 v0 = sv[0], v1 = sv[1], v2 = sv[2], v3 = sv[3];
    float w0 = v0.x, w1 = v0.y, w2 = v0.z, w3 = v0.w;
    float w4 = v1.x, w5 = v1.y, w6 = v1.z, w7 = v1.w;
    float w8 = v2.x, w9 = v2.y, w10 = v2.z, w11 = v2.w;
    float w12 = v3.x, w13 = v3.y, w14 = v3.z, w15 = v3.w;

    float o0 = med9(w0, w1, w2, w3, w4, w5, w6, w7, w8);
    float o1 = med9(w1, w2, w3, w4, w5, w6, w7, w8, w9);
    float o2 = med9(w2, w3, w4, w5, w6, w7, w8, w9, w10);
    float o3 = med9(w3, w4, w5, w6, w7, w8, w9, w10, w11);
    float o4 = med9(w4, w5, w6, w7, w8, w9, w10, w11, w12);
    float o5 = med9(w5, w6, w7, w8, w9, w10, w11, w12, w13);
    float o6 = med9(w6, w7, w8, w9, w10, w11, w12, w13, w14);
    float o7 = med9(w7, w8, w9, w10, w11, w12, w13, w14, w15);

    float4* dst = (float4*)(y + rowOff + s + lane * OPL);   // 16B aligned
    dst[0] = make_float4(o0, o1, o2, o3);
    dst[1] = make_float4(o4, o5, o6, o7);
}

extern "C" void kernel_launch(void* const* d_in, const int* in_sizes, int n_in,
                              void* d_out, int out_size, void* d_ws, size_t ws_size,
                              hipStream_t stream) {
    (void)in_sizes; (void)n_in; (void)d_ws; (void)ws_size;
    const float* x = (const float*)d_in[0];
    float* y = (float*)d_out;

    const int L = 131072;                 // reference row length (fixed)
    const int rows = out_size / L;        // B*C = 256
    dim3 grid(L / OPB, rows);             // (64, 256)
    MedianFilter1D_kernel<<<grid, LANES * NWAVES, 0, stream>>>(x, y, L);
}